// GCNPredictionNet2_13297218748541
// MI455X (gfx1250) — compile-verified
//
#include <hip/hip_runtime.h>
#include <hip/hip_bf16.h>

#define N_NODES 100000

typedef __attribute__((ext_vector_type(2))) float v2f;
typedef __attribute__((ext_vector_type(8))) float v8f;

__device__ __forceinline__ void atomAddF(float* p, float v) {
    unsafeAtomicAdd(p, v);   // lowers to global_atomic_add_f32 on gfx1250
}

__device__ __forceinline__ v8f wmma_f32(v2f a, v2f b, v8f c) {
    // V_WMMA_F32_16X16X4_F32 : D(16x16 f32) = A(16x4 f32) x B(4x16 f32) + C
    return __builtin_amdgcn_wmma_f32_16x16x4_f32(
        /*neg_a=*/false, a, /*neg_b=*/false, b,
        /*c_mod=*/(short)0, c, /*reuse_a=*/false, /*reuse_b=*/false);
}

// ---------------- K0: zero workspace ----------------
__global__ void zero_kernel(float* __restrict__ p, size_t n) {
    size_t i = (size_t)blockIdx.x * blockDim.x + threadIdx.x;
    if (i < n) p[i] = 0.0f;
}

// ---------------- K1: agg1[dst] += x[src], 16 feats, 4 lanes/edge ----------------
__global__ void scatter16_kernel(const float* __restrict__ x,
                                 const int* __restrict__ ei,
                                 float* __restrict__ agg1, int E) {
    int tid = blockIdx.x * blockDim.x + threadIdx.x;
    if (tid >= E * 4) return;
    int e = tid >> 2, part = tid & 3;
    int src = ei[e];
    int dst = ei[E + e];
    const float4 v = *(const float4*)(x + (size_t)src * 16 + part * 4);
    float* p = agg1 + (size_t)dst * 16 + part * 4;
    atomAddF(p + 0, v.x); atomAddF(p + 1, v.y);
    atomAddF(p + 2, v.z); atomAddF(p + 3, v.w);
}

// ---------------- K2: h1 = relu(agg1@W_rel1 + x@W_root1 + b_rel1) ----------------
__global__ __launch_bounds__(32) void conv1_kernel(
    const float* __restrict__ x, const float* __restrict__ agg1,
    const float* __restrict__ W_rel1, const float* __restrict__ b_rel1,
    const float* __restrict__ W_root1, float* __restrict__ h1) {
    const int base = blockIdx.x * 16;
    const int lane = threadIdx.x;
    const int hf   = lane >> 4;      // half-wave select
    const int col  = lane & 15;      // N for B/C/D, M for A loads
    const int colc = col & 7;        // clamped column: safe address for lanes 8..15
    const float msk = (col < 8) ? 1.0f : 0.0f;  // zero-pad mask, branch-free
    v8f c = {};
#pragma unroll
    for (int kk = 0; kk < 16; kk += 4) {
        const int k0 = kk + 2 * hf;
        v2f a_agg = { agg1[(size_t)(base + col) * 16 + k0],
                      agg1[(size_t)(base + col) * 16 + k0 + 1] };
        v2f a_x   = { x[(size_t)(base + col) * 16 + k0],
                      x[(size_t)(base + col) * 16 + k0 + 1] };
        // W_rel1/W_root1 are [16,8]; pad N to 16 via mask (no divergent loads)
        v2f b_rel  = { W_rel1[k0 * 8 + colc] * msk,
                       W_rel1[(k0 + 1) * 8 + colc] * msk };
        v2f b_root = { W_root1[k0 * 8 + colc] * msk,
                       W_root1[(k0 + 1) * 8 + colc] * msk };
        c = wmma_f32(a_agg, b_rel, c);
        c = wmma_f32(a_x, b_root, c);
    }
    if (col < 8) {
        const float bias = b_rel1[col];
#pragma unroll
        for (int g = 0; g < 8; ++g) {
            float v = c[g] + bias;
            v = fmaxf(v, 0.0f);
            h1[(size_t)(base + g + 8 * hf) * 8 + col] = v;   // [N,8]
        }
    }
}

// ---------------- K3: agg2[dst] += h1[src], 8 feats, 2 lanes/edge ----------------
__global__ void scatter8_kernel(const float* __restrict__ h1,
                                const int* __restrict__ ei,
                                float* __restrict__ agg2, int E) {
    int tid = blockIdx.x * blockDim.x + threadIdx.x;
    if (tid >= E * 2) return;
    int e = tid >> 1, part = tid & 1;
    int src = ei[e];
    int dst = ei[E + e];
    const float4 v = *(const float4*)(h1 + (size_t)src * 8 + part * 4);
    float* p = agg2 + (size_t)dst * 8 + part * 4;
    atomAddF(p + 0, v.x); atomAddF(p + 1, v.y);
    atomAddF(p + 2, v.z); atomAddF(p + 3, v.w);
}

// ---------------- K4: h2=relu(conv2); h3=relu(h2@W_fc1+b); out=h3@W_fc2+b ----------------
__global__ __launch_bounds__(32) void conv2_fc_kernel(
    const float* __restrict__ h1, const float* __restrict__ agg2,
    const float* __restrict__ W_rel2, const float* __restrict__ b_rel2,
    const float* __restrict__ W_root2,
    const float* __restrict__ W_fc1, const float* __restrict__ b_fc1,
    const float* __restrict__ W_fc2, const float* __restrict__ b_fc2,
    float* __restrict__ out, float* __restrict__ sum_buf) {
    __shared__ float lds[16 * 17];
    const int base = blockIdx.x * 16;
    const int lane = threadIdx.x;
    const int hf   = lane >> 4;
    const int col  = lane & 15;

    // conv2: C(16x16) = agg2(16x8)@W_rel2(8x16) + h1(16x8)@W_root2(8x16)
    v8f c = {};
#pragma unroll
    for (int kk = 0; kk < 8; kk += 4) {
        const int k0 = kk + 2 * hf;
        v2f a_agg = { agg2[(size_t)(base + col) * 8 + k0],
                      agg2[(size_t)(base + col) * 8 + k0 + 1] };
        v2f a_h   = { h1[(size_t)(base + col) * 8 + k0],
                      h1[(size_t)(base + col) * 8 + k0 + 1] };
        v2f b_rel  = { W_rel2[k0 * 16 + col], W_rel2[(k0 + 1) * 16 + col] };
        v2f b_root = { W_root2[k0 * 16 + col], W_root2[(k0 + 1) * 16 + col] };
        c = wmma_f32(a_agg, b_rel, c);
        c = wmma_f32(a_h, b_root, c);
    }
    const float bias2 = b_rel2[col];
#pragma unroll
    for (int g = 0; g < 8; ++g) {
        float v = fmaxf(c[g] + bias2, 0.0f);           // h2 element (M=g+8*hf, N=col)
        lds[(g + 8 * hf) * 17 + col] = v;              // stage for A-layout reload
    }
    __syncthreads();

    // fc1: h3(16x32) = h2(16x16) @ W_fc1(16x32), two N-blocks of 16
    v8f c0 = {}, c1 = {};
#pragma unroll
    for (int kk = 0; kk < 16; kk += 4) {
        const int k0 = kk + 2 * hf;
        v2f a  = { lds[col * 17 + k0], lds[col * 17 + k0 + 1] };
        v2f b0 = { W_fc1[k0 * 32 + col],      W_fc1[(k0 + 1) * 32 + col] };
        v2f b1 = { W_fc1[k0 * 32 + 16 + col], W_fc1[(k0 + 1) * 32 + 16 + col] };
        c0 = wmma_f32(a, b0, c0);
        c1 = wmma_f32(a, b1, c1);
    }
    const float bf0 = b_fc1[col], bf1 = b_fc1[16 + col];
    const float w2a = W_fc2[col], w2b = W_fc2[16 + col];
    const float bout = b_fc2[0];
    float lsum = 0.0f;
#pragma unroll
    for (int g = 0; g < 8; ++g) {
        float v0 = fmaxf(c0[g] + bf0, 0.0f);
        float v1 = fmaxf(c1[g] + bf1, 0.0f);
        float p = v0 * w2a + v1 * w2b;       // fc2 partial over this lane's 2 cols
        p += __shfl_xor(p, 1, 16);           // butterfly reduce across 16 cols
        p += __shfl_xor(p, 2, 16);
        p += __shfl_xor(p, 4, 16);
        p += __shfl_xor(p, 8, 16);
        const float oval = p + bout;
        if (col == 0) {
            out[base + g + 8 * hf] = oval;
            lsum += oval;
        }
    }
    const float other = __shfl_xor(lsum, 16, 32);      // lane0 <- lane16's 8 rows
    if (lane == 0) atomAddF(sum_buf, lsum + other);
}

// ---------------- K5: out -= mean ----------------
__global__ void mean_sub_kernel(float* __restrict__ out,
                                const float* __restrict__ sum_buf, int n) {
    int i = blockIdx.x * blockDim.x + threadIdx.x;
    if (i < n) out[i] -= sum_buf[0] * (1.0f / (float)n);
}

extern "C" void kernel_launch(void* const* d_in, const int* in_sizes, int n_in,
                              void* d_out, int out_size, void* d_ws, size_t ws_size,
                              hipStream_t stream) {
    const float* x       = (const float*)d_in[0];
    const int*   ei      = (const int*)d_in[1];
    const float* W_rel1  = (const float*)d_in[2];
    const float* b_rel1  = (const float*)d_in[3];
    const float* W_root1 = (const float*)d_in[4];
    const float* W_rel2  = (const float*)d_in[5];
    const float* b_rel2  = (const float*)d_in[6];
    const float* W_root2 = (const float*)d_in[7];
    const float* W_fc1   = (const float*)d_in[8];
    const float* b_fc1   = (const float*)d_in[9];
    const float* W_fc2   = (const float*)d_in[10];
    const float* b_fc2   = (const float*)d_in[11];
    float* out = (float*)d_out;
    float* ws  = (float*)d_ws;

    const int N = N_NODES;
    const int E = in_sizes[1] / 2;

    float* agg1 = ws;                        // [N,16]
    float* h1   = ws + (size_t)N * 16;       // [N,8]
    float* agg2 = ws + (size_t)N * 24;       // [N,8]
    float* sumb = ws + (size_t)N * 32;       // scalar

    const size_t zn = (size_t)N * 32 + 8;    // zero agg1,h1,agg2,sum
    zero_kernel<<<(unsigned)((zn + 511) / 512), 512, 0, stream>>>(ws, zn);

    scatter16_kernel<<<(unsigned)(((size_t)E * 4 + 255) / 256), 256, 0, stream>>>(x, ei, agg1, E);

    conv1_kernel<<<N / 16, 32, 0, stream>>>(x, agg1, W_rel1, b_rel1, W_root1, h1);

    scatter8_kernel<<<(unsigned)(((size_t)E * 2 + 255) / 256), 256, 0, stream>>>(h1, ei, agg2, E);

    conv2_fc_kernel<<<N / 16, 32, 0, stream>>>(h1, agg2, W_rel2, b_rel2, W_root2,
                                               W_fc1, b_fc1, W_fc2, b_fc2, out, sumb);

    mean_sub_kernel<<<(N + 255) / 256, 256, 0, stream>>>(out, sumb, N);
}